// SRWMlayer_14851996910186
// MI455X (gfx1250) — compile-verified
//
#include <hip/hip_runtime.h>
#include <math.h>

typedef __attribute__((ext_vector_type(2))) float v2f;
typedef __attribute__((ext_vector_type(8))) float v8f;
typedef int v4i __attribute__((vector_size(16)));   // matches builtin param type

#define SLEN   256
#define BSZ    8
#define NHEAD  16
#define DHEAD  64
#define INDIM  1024
#define LN_EPS 1e-5f
#define WPAD   65   // padded row stride for 64x64 LDS tiles (bank-conflict free)

// GEMM tiling
#define BM   128
#define BN   64
#define BK   16
#define APAD 20     // row pad: multiple of 4 (16B-aligned rows) and conflict-free

#if defined(__gfx1250__) && __has_builtin(__builtin_amdgcn_global_load_async_to_lds_b128)
#define HAVE_ASYNC_LDS 1
typedef __attribute__((address_space(1))) v4i* as1_v4i_ptr;
typedef __attribute__((address_space(3))) v4i* as3_v4i_ptr;
#else
#define HAVE_ASYNC_LDS 0
#endif

__device__ __forceinline__ void wait_async_lds() {
#if defined(__gfx1250__)
#if __has_builtin(__builtin_amdgcn_s_wait_asynccnt)
  __builtin_amdgcn_s_wait_asynccnt(0);
#else
  asm volatile("s_wait_asynccnt 0x0" ::: "memory");
#endif
#endif
}

// ---------------------------------------------------------------------------
// Kernel 1: SRWM sequential scan. One workgroup per (batch, head) cell.
// Fast-weight state lives in LDS; rank-1 updates use V_WMMA_F32_16X16X4_F32.
// ---------------------------------------------------------------------------
__global__ __launch_bounds__(256) void srwm_scan_kernel(
    const float* __restrict__ h,     // (SLEN,BSZ,INDIM)
    const float* __restrict__ W_y0,  // (1,NHEAD,64,64)
    const float* __restrict__ W_q0,
    const float* __restrict__ W_k0,
    const float* __restrict__ w_b0,  // (1,NHEAD,64,4)
    float* __restrict__ ys)          // (SLEN,BSZ,INDIM)
{
  __shared__ float sW[3 * 64 * WPAD];   // Wy | Wq | Wk
  __shared__ float swb[64 * 4];
  __shared__ float sx[64];
  __shared__ float sv[192];             // y | q_raw | k_raw
  __shared__ float sq[64], sk[64], sqmk[64];
  __shared__ float sdv[192];
  __shared__ float sbin[4], sbeta[4], sdvb[4];
  __shared__ float sred[4];

  const int tid  = threadIdx.x;
  const int cell = blockIdx.x;          // 0..127
  const int b    = cell / NHEAD;
  const int hd   = cell % NHEAD;

  // Load initial fast weights (broadcast over batch) into padded LDS tiles.
  const float* Wsrc0 = W_y0 + hd * 64 * 64;
  const float* Wsrc1 = W_q0 + hd * 64 * 64;
  const float* Wsrc2 = W_k0 + hd * 64 * 64;
  for (int i = tid; i < 64 * 64; i += 256) {
    int r = i >> 6, c = i & 63;
    sW[0 * 64 * WPAD + r * WPAD + c] = Wsrc0[i];
    sW[1 * 64 * WPAD + r * WPAD + c] = Wsrc1[i];
    sW[2 * 64 * WPAD + r * WPAD + c] = Wsrc2[i];
  }
  swb[tid] = w_b0[hd * 256 + tid];
  __syncthreads();

  const int lane = tid & 31;            // wave32
  const int wave = tid >> 5;            // 8 waves
  const int lrow = lane & 15;
  const int lhi  = lane >> 4;

  for (int t = 0; t < SLEN; ++t) {
    // ---- 1. load x_t ----
    const float* xptr = h + ((size_t)t * BSZ + b) * INDIM + hd * DHEAD;
    if (tid < 64) sx[tid] = xptr[tid];
    __syncthreads();

    // ---- 2. matvecs with W_{t-1}: y, q_raw, k_raw, beta_in ----
    if (tid < 192) {
      int m = tid >> 6, r = tid & 63;
      const float* Wr = &sW[m * 64 * WPAD + r * WPAD];
      float acc = 0.f;
      #pragma unroll 8
      for (int j = 0; j < 64; ++j) acc += Wr[j] * sx[j];
      sv[tid] = acc;
    } else if (tid < 196) {
      int c = tid - 192;
      float acc = 0.f;
      for (int j = 0; j < 64; ++j) acc += swb[j * 4 + c] * sx[j];
      sbin[c] = acc;
    }
    __syncthreads();

    // ---- 3. softmax(q), softmax(k), sigmoid(beta) ----
    if (tid < 2) {
      const float* vv = &sv[64 + tid * 64];
      float mx = vv[0];
      for (int j = 1; j < 64; ++j) mx = fmaxf(mx, vv[j]);
      sred[tid] = mx;
    }
    __syncthreads();
    if (tid < 64)        sq[tid]      = __expf(sv[64 + tid] - sred[0]);
    else if (tid < 128)  sk[tid - 64] = __expf(sv[128 + (tid - 64)] - sred[1]);
    __syncthreads();
    if (tid < 2) {
      const float* vv = (tid == 0) ? sq : sk;
      float s = 0.f;
      for (int j = 0; j < 64; ++j) s += vv[j];
      sred[2 + tid] = s;
    }
    __syncthreads();
    if (tid < 64) {
      float qn = sq[tid] / sred[2];
      float kn = sk[tid] / sred[3];
      sq[tid] = qn; sk[tid] = kn; sqmk[tid] = qn - kn;
    } else if (tid < 68) {
      int c = tid - 64;
      sbeta[c] = 1.f / (1.f + __expf(-sbin[c]));
    }
    // ---- 4. emit y_t (computed from W_{t-1}) ----
    if (tid < 64)
      ys[((size_t)t * BSZ + b) * INDIM + hd * DHEAD + tid] = sv[tid];
    __syncthreads();

    // ---- 5. dv = W_{t-1} (q-k), dvb = wb^T (q-k) ----
    if (tid < 192) {
      int m = tid >> 6, r = tid & 63;
      const float* Wr = &sW[m * 64 * WPAD + r * WPAD];
      float acc = 0.f;
      #pragma unroll 8
      for (int j = 0; j < 64; ++j) acc += Wr[j] * sqmk[j];
      sdv[tid] = acc;
    } else if (tid < 196) {
      int c = tid - 192;
      float acc = 0.f;
      for (int j = 0; j < 64; ++j) acc += swb[j * 4 + c] * sqmk[j];
      sdvb[c] = acc;
    }
    __syncthreads();

    // ---- 6a. W += beta * dv k^T  via f32 WMMA (K padded 1 -> 4) ----
    // 3 matrices * 16 (16x16) blocks = 48 blocks; 6 per wave, 8 waves.
    #pragma unroll
    for (int i = 0; i < 6; ++i) {
      int blk = wave * 6 + i;                 // 0..47, uniform per wave
      int m   = blk >> 4;                     // which matrix
      int sub = blk & 15;
      int m0  = (sub >> 2) << 4;
      int n0  = (sub & 3) << 4;
      float* Wt  = &sW[m * 64 * WPAD];
      float  bta = sbeta[m];
      // A (16x4): col K=0 = beta*dv block; lanes 0-15 carry K=0/1, 16-31 K=2/3
      v2f A; A.x = (lhi == 0) ? bta * sdv[m * 64 + m0 + lrow] : 0.f; A.y = 0.f;
      // B (4x16): row K=0 = k block
      v2f B; B.x = (lhi == 0) ? sk[n0 + lrow] : 0.f;                 B.y = 0.f;
      v8f C;
      #pragma unroll
      for (int v = 0; v < 8; ++v)
        C[v] = Wt[(m0 + v + 8 * lhi) * WPAD + n0 + lrow];
      v8f D = __builtin_amdgcn_wmma_f32_16x16x4_f32(
          false, A, false, B, (short)0, C, false, false);
      #pragma unroll
      for (int v = 0; v < 8; ++v)
        Wt[(m0 + v + 8 * lhi) * WPAD + n0 + lrow] = D[v];
    }
    // ---- 6b. wb += beta3 * k dvb^T (one element per thread) ----
    {
      int j = tid >> 2, c = tid & 3;
      swb[j * 4 + c] += sbeta[3] * sk[j] * sdvb[c];
    }
    __syncthreads();
  }
}

// ---------------------------------------------------------------------------
// Kernel 2: per-row LayerNorm statistics (mu, rstd) for h. One block per row.
// ---------------------------------------------------------------------------
__global__ __launch_bounds__(256) void ln_stats_kernel(
    const float* __restrict__ h, float* __restrict__ mu, float* __restrict__ rstd)
{
  __shared__ float s1[256], s2[256];
  const int row = blockIdx.x;
  const float* hp = h + (size_t)row * INDIM;
  float a = 0.f, b = 0.f;
  for (int j = threadIdx.x; j < INDIM; j += 256) {
    float v = hp[j]; a += v; b += v * v;
  }
  s1[threadIdx.x] = a; s2[threadIdx.x] = b;
  __syncthreads();
  for (int off = 128; off > 0; off >>= 1) {
    if (threadIdx.x < off) {
      s1[threadIdx.x] += s1[threadIdx.x + off];
      s2[threadIdx.x] += s2[threadIdx.x + off];
    }
    __syncthreads();
  }
  if (threadIdx.x == 0) {
    float m   = s1[0] * (1.f / INDIM);
    float var = s2[0] * (1.f / INDIM) - m * m;
    mu[row]   = m;
    rstd[row] = rsqrtf(var + LN_EPS);
  }
}

// ---------------------------------------------------------------------------
// Kernel 3: out = ys @ out_w^T + LN(h)*gamma + beta.
// LDS-staged 128x64 block tile; async global->LDS staging when available.
// Each wave computes a 16x64 strip (4 f32-WMMA accumulators, A reused 4x).
// ---------------------------------------------------------------------------
__global__ __launch_bounds__(256) void out_gemm_ln_kernel(
    const float* __restrict__ ys,     // (M=2048, K=1024)
    const float* __restrict__ out_w,  // (N=1024, K=1024) row-major
    const float* __restrict__ h,
    const float* __restrict__ mu,
    const float* __restrict__ rstd,
    const float* __restrict__ gamma,
    const float* __restrict__ beta,
    float* __restrict__ out)          // (M, N)
{
  __shared__ __align__(16) float sA[BM * APAD];   // 128 x 16 (padded)
  __shared__ __align__(16) float sB[BN * APAD];   //  64 x 16 (padded)

  const int tid  = threadIdx.x;
  const int lane = tid & 31;
  const int wave = tid >> 5;
  const int lrow = lane & 15;
  const int lhi  = lane >> 4;

  const int mBlk = (blockIdx.x >> 4) * BM;   // 16 m-blocks
  const int nBlk = (blockIdx.x & 15) * BN;   // 16 n-blocks

  // Cooperative staging map: A -> 2 threads/row x 8 floats, B -> 4 threads/row x 4 floats
  const int arow = tid >> 1, acol = (tid & 1) * 8;
  const int brow = tid >> 2, bcol = (tid & 3) * 4;

  v8f acc[4] = {};

  for (int k0 = 0; k0 < INDIM; k0 += BK) {
    const float* ga = ys    + (size_t)(mBlk + arow) * INDIM + k0 + acol;
    const float* gb = out_w + (size_t)(nBlk + brow) * INDIM + k0 + bcol;
    float* la = &sA[arow * APAD + acol];
    float* lb = &sB[brow * APAD + bcol];

#if HAVE_ASYNC_LDS
    // gfx1250 async global->LDS (ASYNCcnt path): two B128 for A, one for B.
    __builtin_amdgcn_global_load_async_to_lds_b128(
        (as1_v4i_ptr)ga,       (as3_v4i_ptr)la,       0, 0);
    __builtin_amdgcn_global_load_async_to_lds_b128(
        (as1_v4i_ptr)(ga + 4), (as3_v4i_ptr)(la + 4), 0, 0);
    __builtin_amdgcn_global_load_async_to_lds_b128(
        (as1_v4i_ptr)gb,       (as3_v4i_ptr)lb,       0, 0);
    if (k0 + BK < INDIM) {              // prefetch next K tile into cache
      __builtin_prefetch(ga + BK, 0, 0);
      __builtin_prefetch(gb + BK, 0, 0);
    }
    wait_async_lds();
    __syncthreads();
#else
    float4 a0 = *(const float4*)ga;
    float4 a1 = *(const float4*)(ga + 4);
    float4 b0 = *(const float4*)gb;
    if (k0 + BK < INDIM) {
      __builtin_prefetch(ga + BK, 0, 0);
      __builtin_prefetch(gb + BK, 0, 0);
    }
    *(float4*)la       = a0;
    *(float4*)(la + 4) = a1;
    *(float4*)lb       = b0;
    __syncthreads();
#endif

    // Compute: 4 K-chunks of 4, 4 N-tiles per wave.
    #pragma unroll
    for (int kk = 0; kk < BK; kk += 4) {
      const float* ap = &sA[(wave * 16 + lrow) * APAD + kk + 2 * lhi];
      v2f A; A.x = ap[0]; A.y = ap[1];
      #pragma unroll
      for (int nt = 0; nt < 4; ++nt) {
        const float* bp = &sB[(nt * 16 + lrow) * APAD + kk + 2 * lhi];
        v2f B; B.x = bp[0]; B.y = bp[1];
        acc[nt] = __builtin_amdgcn_wmma_f32_16x16x4_f32(
            false, A, false, B, (short)0, acc[nt], false, false);
      }
    }
    __syncthreads();
  }

  // Fused LayerNorm + residual epilogue; wave writes its 16x64 strip.
  #pragma unroll
  for (int nt = 0; nt < 4; ++nt) {
    #pragma unroll
    for (int v = 0; v < 8; ++v) {
      int m = mBlk + wave * 16 + v + 8 * lhi;
      int n = nBlk + nt * 16 + lrow;
      size_t idx = (size_t)m * INDIM + n;
      float hv  = h[idx];
      float lnv = (hv - mu[m]) * rstd[m] * gamma[n] + beta[n];
      out[idx] = acc[nt][v] + lnv;
    }
  }
}

// ---------------------------------------------------------------------------
extern "C" void kernel_launch(void* const* d_in, const int* in_sizes, int n_in,
                              void* d_out, int out_size, void* d_ws, size_t ws_size,
                              hipStream_t stream) {
  const float* h     = (const float*)d_in[0];
  const float* W_y   = (const float*)d_in[1];
  const float* W_q   = (const float*)d_in[2];
  const float* W_k   = (const float*)d_in[3];
  const float* w_b   = (const float*)d_in[4];
  const float* out_w = (const float*)d_in[5];
  const float* gamma = (const float*)d_in[6];
  const float* beta  = (const float*)d_in[7];
  float* out = (float*)d_out;

  // Workspace layout: ys (2048*1024 f32) | mu (2048) | rstd (2048)
  float* ys   = (float*)d_ws;
  float* mu   = ys + (size_t)SLEN * BSZ * INDIM;
  float* rstd = mu + SLEN * BSZ;

  srwm_scan_kernel<<<BSZ * NHEAD, 256, 0, stream>>>(h, W_y, W_q, W_k, w_b, ys);
  ln_stats_kernel<<<SLEN * BSZ, 256, 0, stream>>>(h, mu, rstd);
  // (2048/128) x (1024/64) = 256 blocks, 256 threads (8 waves) each
  out_gemm_ln_kernel<<<256, 256, 0, stream>>>(ys, out_w, h, mu, rstd,
                                              gamma, beta, out);
}